// topkBCE_35880156791412
// MI455X (gfx1250) — compile-verified
//
#include <hip/hip_runtime.h>

#define LOG_CLAMP  (-100.0f)
#define NEG_RATIO  (3.0f)
#define EPS_F      (1e-4f)

typedef __attribute__((ext_vector_type(2))) float v2f;
typedef __attribute__((ext_vector_type(8))) float v8f;

struct TopkCtl {
  unsigned pos_count;     // atomic u32
  unsigned neg_count;     // atomic u32
  float    positive_sum;  // atomic f32
  unsigned k;
  unsigned prefix;        // running selected key prefix
  unsigned cnt_above;     // count strictly above current refined region
  float    sum_above;     // sum strictly above current refined region
  unsigned pad_;
};

#if defined(__AMDGCN__) && __has_builtin(__builtin_amdgcn_wmma_f32_16x16x4_f32)
#define USE_WMMA_REDUCE 1
#else
#define USE_WMMA_REDUCE 0
#endif

// Reduce three per-lane scalars across the full wave32.
// WMMA trick: A(16x4) with VGPR0=x, VGPR1=y so A[m,0]=x_m, A[m,1]=y_m (lanes 0-15)
// and A[m,2]=x_{m+16}, A[m,3]=y_{m+16} (lanes 16-31). B(4x16) with VGPR0=1,VGPR1=0
// selects K rows {0,2}: D[m,n] = x_m + x_{m+16}. Per-lane sum of the 8 D VGPRs gives
// each 16-lane half's row-sum; one shfl_xor(16) completes the 32-lane total.
__device__ __forceinline__ void waveSum3(float& x, float& y, float& z) {
#if USE_WMMA_REDUCE
  v2f a;   a[0] = x;    a[1] = y;
  v2f b02; b02[0] = 1.0f; b02[1] = 0.0f;   // K rows 0,2 = 1
  v2f b13; b13[0] = 0.0f; b13[1] = 1.0f;   // K rows 1,3 = 1
  v8f c = {};
  v8f dx = __builtin_amdgcn_wmma_f32_16x16x4_f32(false, a, false, b02, (short)0, c, false, false);
  v8f dy = __builtin_amdgcn_wmma_f32_16x16x4_f32(false, a, false, b13, (short)0, c, false, false);
  v2f a2; a2[0] = z; a2[1] = 0.0f;
  v8f dz = __builtin_amdgcn_wmma_f32_16x16x4_f32(false, a2, false, b02, (short)0, c, false, false);
  float hx = dx[0]+dx[1]+dx[2]+dx[3]+dx[4]+dx[5]+dx[6]+dx[7];
  float hy = dy[0]+dy[1]+dy[2]+dy[3]+dy[4]+dy[5]+dy[6]+dy[7];
  float hz = dz[0]+dz[1]+dz[2]+dz[3]+dz[4]+dz[5]+dz[6]+dz[7];
  x = hx + __shfl_xor(hx, 16);
  y = hy + __shfl_xor(hy, 16);
  z = hz + __shfl_xor(hz, 16);
#else
  for (int m = 16; m > 0; m >>= 1) {
    x += __shfl_xor(x, m);
    y += __shfl_xor(y, m);
    z += __shfl_xor(z, m);
  }
#endif
}

__device__ __forceinline__ float bce_loss(float g, float p) {
  float lp = fmaxf(__logf(p),        LOG_CLAMP);
  float lq = fmaxf(__logf(1.0f - p), LOG_CLAMP);
  return -(g * lp + (1.0f - g) * lq);
}

// ---------------- Pass A: loss + masks + level-0 histogram (top 12 key bits) ---
__global__ void __launch_bounds__(256) topk_passA(
    const float* __restrict__ gt, const float* __restrict__ pred,
    unsigned n4, unsigned n,
    TopkCtl* ctl, unsigned* __restrict__ h0c, float* __restrict__ h0s,
    unsigned* __restrict__ keys /* nullable */) {
  __shared__ unsigned sc[4096];
  __shared__ float    ss[4096];
  __shared__ float    red[3][8];
  for (unsigned i = threadIdx.x; i < 4096u; i += 256u) { sc[i] = 0u; ss[i] = 0.0f; }
  __syncthreads();

  float posCnt = 0.0f, posSum = 0.0f, negCnt = 0.0f;
  const float4* g4 = (const float4*)gt;
  const float4* p4 = (const float4*)pred;
  unsigned stride = gridDim.x * blockDim.x;

  for (unsigned i = blockIdx.x * blockDim.x + threadIdx.x; i < n4; i += stride) {
    __builtin_prefetch(g4 + i + stride, 0, 1);   // -> global_prefetch_b8
    __builtin_prefetch(p4 + i + stride, 0, 1);
    float4 g = g4[i];
    float4 p = p4[i];
    float gv[4] = {g.x, g.y, g.z, g.w};
    float pv[4] = {p.x, p.y, p.z, p.w};
    unsigned kv[4];
#pragma unroll
    for (int j = 0; j < 4; ++j) {
      float gg = gv[j], pp = pv[j];
      float loss = bce_loss(gg, pp);
      bool pos = gg >= 0.9f;
      bool ign = (!pos) && (gg >= 0.8f);
      unsigned key = 0xFFFFFFFFu;                // sentinel: not a negative
      if (pos) { posCnt += 1.0f; posSum += loss; }
      else if (!ign) {
        negCnt += 1.0f;
        key = __float_as_uint(loss);             // loss >= 0 -> bits are order-preserving
        atomicAdd(&sc[key >> 20], 1u);           // ds_add_u32
        atomicAdd(&ss[key >> 20], loss);         // ds_add_f32
      }
      kv[j] = key;
    }
    if (keys) {
      uint4 kk; kk.x = kv[0]; kk.y = kv[1]; kk.z = kv[2]; kk.w = kv[3];
      ((uint4*)keys)[i] = kk;
    }
  }

  // scalar tail (n % 4) handled by block 0
  unsigned tailBase = n4 * 4u;
  if (blockIdx.x == 0 && threadIdx.x < (n - tailBase)) {
    unsigned idx = tailBase + threadIdx.x;
    float gg = gt[idx], pp = pred[idx];
    float loss = bce_loss(gg, pp);
    bool pos = gg >= 0.9f;
    bool ign = (!pos) && (gg >= 0.8f);
    unsigned key = 0xFFFFFFFFu;
    if (pos) { posCnt += 1.0f; posSum += loss; }
    else if (!ign) {
      negCnt += 1.0f;
      key = __float_as_uint(loss);
      atomicAdd(&sc[key >> 20], 1u);
      atomicAdd(&ss[key >> 20], loss);
    }
    if (keys) keys[idx] = key;
  }

  __syncthreads();
  for (unsigned i = threadIdx.x; i < 4096u; i += 256u) {
    unsigned c = sc[i];
    if (c) { atomicAdd(&h0c[i], c); atomicAdd(&h0s[i], ss[i]); }
  }

  // cross-lane reduce (WMMA path) then block combine; EXEC is all-1s here
  waveSum3(posCnt, posSum, negCnt);
  unsigned lane = threadIdx.x & 31u, wave = threadIdx.x >> 5;
  if (lane == 0) { red[0][wave] = posCnt; red[1][wave] = posSum; red[2][wave] = negCnt; }
  __syncthreads();
  if (threadIdx.x == 0) {
    float p0 = 0.f, p1 = 0.f, p2 = 0.f;
    for (int w = 0; w < 8; ++w) { p0 += red[0][w]; p1 += red[1][w]; p2 += red[2][w]; }
    if (p0 != 0.f) atomicAdd(&ctl->pos_count, (unsigned)(p0 + 0.5f));
    if (p1 != 0.f) atomicAdd(&ctl->positive_sum, p1);
    if (p2 != 0.f) atomicAdd(&ctl->neg_count, (unsigned)(p2 + 0.5f));
  }
}

// ---------------- Refinement histogram passes (level 1: mid 12 bits, level 2: low 8) ---
__global__ void __launch_bounds__(256) topk_refine(
    const float* __restrict__ gt, const float* __restrict__ pred,
    const unsigned* __restrict__ keys /* nullable */, unsigned n,
    const TopkCtl* __restrict__ ctl,
    unsigned* __restrict__ hc, float* __restrict__ hs, int level) {
  __shared__ unsigned sc[4096];
  __shared__ float    ss[4096];
  unsigned nb = (level == 1) ? 4096u : 256u;
  for (unsigned i = threadIdx.x; i < nb; i += 256u) { sc[i] = 0u; ss[i] = 0.0f; }
  __syncthreads();
  unsigned prefix = ctl->prefix;
  unsigned stride = gridDim.x * blockDim.x;
  for (unsigned i = blockIdx.x * blockDim.x + threadIdx.x; i < n; i += stride) {
    unsigned key;
    if (keys) {
      __builtin_prefetch(keys + i + stride, 0, 1);
      key = keys[i];
      if (key == 0xFFFFFFFFu) continue;
    } else {
      float gg = gt[i];
      bool pos = gg >= 0.9f;
      bool ign = (!pos) && (gg >= 0.8f);
      if (pos || ign) continue;
      key = __float_as_uint(bce_loss(gg, pred[i]));
    }
    if (level == 1) {
      if ((key >> 20) == prefix) {
        unsigned b = (key >> 8) & 0xFFFu;
        atomicAdd(&sc[b], 1u);
        atomicAdd(&ss[b], __uint_as_float(key));
      }
    } else {
      if ((key >> 8) == prefix) {
        unsigned b = key & 0xFFu;
        atomicAdd(&sc[b], 1u);
        atomicAdd(&ss[b], __uint_as_float(key));
      }
    }
  }
  __syncthreads();
  for (unsigned i = threadIdx.x; i < nb; i += 256u) {
    unsigned c = sc[i];
    if (c) { atomicAdd(&hc[i], c); atomicAdd(&hs[i], ss[i]); }
  }
}

// ---------------- Scan: pick threshold bin at this level, accumulate above-sums ---
__global__ void __launch_bounds__(1024) topk_scan(
    TopkCtl* ctl, const unsigned* __restrict__ hc, const float* __restrict__ hs,
    int nbins, int level, float* __restrict__ out) {
  __shared__ unsigned ccnt[1024];
  __shared__ float    csum[1024];
  int nch = nbins >> 2;
  if ((int)threadIdx.x < nch) {
    int base = threadIdx.x * 4;
    ccnt[threadIdx.x] = hc[base] + hc[base + 1] + hc[base + 2] + hc[base + 3];
    csum[threadIdx.x] = hs[base] + hs[base + 1] + hs[base + 2] + hs[base + 3];
  }
  __syncthreads();
  if (threadIdx.x != 0) return;

  if (level == 0) {
    unsigned pos = ctl->pos_count;
    unsigned neg = ctl->neg_count;
    // f32 casts on purpose: matches jnp's .astype(float32) semantics
    float kf = fminf(fmaxf((float)pos, 1.0f) * NEG_RATIO, (float)neg);
    ctl->k = (unsigned)floorf(kf);
  }
  unsigned k  = ctl->k;
  unsigned kc = k - ctl->cnt_above;  // remaining needed within this refinement
  unsigned acc = 0; float sacc = 0.0f;
  int b = nbins - 1;
  for (int c = nch - 1; c >= 0; --c) {
    unsigned cc = ccnt[c];
    if (acc + cc >= kc) {
      for (int bb = c * 4 + 3; bb >= c * 4; --bb) {
        unsigned cb = hc[bb];
        if (acc + cb >= kc) { b = bb; goto found; }
        acc += cb; sacc += hs[bb];
      }
    }
    acc += cc; sacc += csum[c];
  }
found:
  ctl->cnt_above += acc;
  ctl->sum_above += sacc;
  unsigned shift = (level == 2) ? 8u : 12u;
  unsigned prefix = (ctl->prefix << shift) | (unsigned)b;
  ctl->prefix = prefix;
  if (level == 2) {
    float v    = __uint_as_float(prefix);  // exact k-th largest value
    float topk = ctl->sum_above + (float)(k - ctl->cnt_above) * v;
    float res  = (NEG_RATIO * ctl->positive_sum + topk)
               / ((float)ctl->pos_count + (float)k + EPS_F);
    out[0] = res;  // WEIGHT == 1.0
  }
}

extern "C" void kernel_launch(void* const* d_in, const int* in_sizes, int n_in,
                              void* d_out, int out_size, void* d_ws, size_t ws_size,
                              hipStream_t stream) {
  const float* gt   = (const float*)d_in[0];   // [B,H,W]   f32
  const float* pred = (const float*)d_in[1];   // [B,1,H,W] f32, same flat length
  unsigned n = (unsigned)in_sizes[0];
  unsigned n4 = n / 4u;

  char* ws = (char*)d_ws;
  TopkCtl*  ctl = (TopkCtl*)ws;
  unsigned* h0c = (unsigned*)(ws + 64);
  float*    h0s = (float*)   (ws + 64 + 1 * 16384);
  unsigned* h1c = (unsigned*)(ws + 64 + 2 * 16384);
  float*    h1s = (float*)   (ws + 64 + 3 * 16384);
  unsigned* h2c = (unsigned*)(ws + 64 + 4 * 16384);
  float*    h2s = (float*)   (ws + 64 + 4 * 16384 + 1024);
  size_t fixedBytes = 64 + 4 * 16384 + 2 * 1024;          // 67,648 B
  size_t keysOff = (fixedBytes + 255) & ~(size_t)255;
  unsigned* keys = nullptr;                                // optional key cache
  if (ws_size >= keysOff + (size_t)n * 4u) keys = (unsigned*)(ws + keysOff);

  (void)hipMemsetAsync(d_ws, 0, fixedBytes, stream);       // zero ctl + histograms

  unsigned blocksA = (n4 + 255u) / 256u;
  if (blocksA > 3072u) blocksA = 3072u;
  if (blocksA == 0u) blocksA = 1u;
  unsigned blocksR = (n + 255u) / 256u;
  if (blocksR > 3072u) blocksR = 3072u;

  topk_passA<<<blocksA, 256, 0, stream>>>(gt, pred, n4, n, ctl, h0c, h0s, keys);
  topk_scan <<<1, 1024, 0, stream>>>(ctl, h0c, h0s, 4096, 0, (float*)d_out);
  topk_refine<<<blocksR, 256, 0, stream>>>(gt, pred, keys, n, ctl, h1c, h1s, 1);
  topk_scan <<<1, 1024, 0, stream>>>(ctl, h1c, h1s, 4096, 1, (float*)d_out);
  topk_refine<<<blocksR, 256, 0, stream>>>(gt, pred, keys, n, ctl, h2c, h2s, 2);
  topk_scan <<<1, 1024, 0, stream>>>(ctl, h2c, h2s, 256, 2, (float*)d_out);
}